// PCKh_36739150250670
// MI455X (gfx1250) — compile-verified
//
#include <hip/hip_runtime.h>
#include <math.h>

// ---------------------------------------------------------------------------
// PCKh metric. x, target: (16, 14, 256, 256) f32. Output: 1 float.
// Bandwidth-bound: stream both tensors once (117 MB) through the gfx1250
// async global->LDS DMA path, reduce with wave32 shuffles.
// ---------------------------------------------------------------------------

#define NB   16
#define NJ   14
#define NH   256
#define NW   256
#define NBJ  (NB * NJ)
#define CH_ROWS 16                    // rows per LDS chunk (16 KB)
#define NCHUNK  (NH / CH_ROWS)

#if defined(__has_builtin)
#if __has_builtin(__builtin_amdgcn_global_load_async_to_lds_b128)
#define USE_ASYNC 1
#endif
#endif
#ifndef USE_ASYNC
#define USE_ASYNC 0
#endif

typedef float f32x4 __attribute__((ext_vector_type(4)));
typedef int   v4i   __attribute__((ext_vector_type(4)));

#if USE_ASYNC
// explicit address-space pointer casts for the async-DMA builtin
#define AS_GLOBAL(p) ((__attribute__((address_space(1))) v4i*)(p))
#define AS_LDS(p)    ((__attribute__((address_space(3))) v4i*)(p))

// ---- chunk fill: 16 rows * 256 cols = 4096 f32 = 1024 x b128 ---------------
__device__ __forceinline__ void issue_chunk_async(const float* __restrict__ g,
                                                  float* l, int tid) {
    const f32x4* gs = (const f32x4*)g;
    f32x4*       ls = (f32x4*)l;
#pragma unroll
    for (int i = 0; i < 4; ++i) {
        int e = tid + (i << 8);
        __builtin_amdgcn_global_load_async_to_lds_b128(AS_GLOBAL(gs + e),
                                                       AS_LDS(ls + e), 0, 0);
    }
}
#else
__device__ __forceinline__ void copy_chunk_sync(const float* __restrict__ g,
                                                float* l, int tid) {
    const f32x4* gs = (const f32x4*)g;
    f32x4*       ls = (f32x4*)l;
#pragma unroll
    for (int i = 0; i < 4; ++i) {
        int e = tid + (i << 8);
        ls[e] = gs[e];
    }
}
#endif

// argmax-combine: prefer larger value; on tie prefer smaller index
__device__ __forceinline__ void amax_combine(float& v0, int& i0, float ov, int oi) {
    if (ov > v0 || (ov == v0 && oi < i0)) { v0 = ov; i0 = oi; }
}

// ---------------------------------------------------------------------------
// Kernel 1: per (b,j) plane -> ys (target row argmax), valid, pred_ys (x).
// ---------------------------------------------------------------------------
__global__ __launch_bounds__(256) void pckh_stats_kernel(
    const float* __restrict__ x, const float* __restrict__ target,
    int* __restrict__ ys_out, int* __restrict__ pred_out, int* __restrict__ valid_out)
{
    __shared__ float buf[2][CH_ROWS * NW];
    __shared__ float wbv[8];
    __shared__ int   wbi[8];

    const int bj   = blockIdx.x;
    const int tid  = threadIdx.x;
    const int wv   = tid >> 5;
    const int lane = tid & 31;

    for (int plane = 0; plane < 2; ++plane) {
        const float* base = (plane == 0 ? target : x) + (size_t)bj * (NH * NW);

        float bestv = -INFINITY;
        int   bestr = 0;

#if USE_ASYNC
        issue_chunk_async(base, buf[0], tid);
#endif
        for (int c = 0; c < NCHUNK; ++c) {
#if USE_ASYNC
            if (c + 1 < NCHUNK) {
                issue_chunk_async(base + (size_t)(c + 1) * CH_ROWS * NW,
                                  buf[(c + 1) & 1], tid);
                asm volatile("s_wait_asynccnt 0x4" ::: "memory");
            } else {
                asm volatile("s_wait_asynccnt 0x0" ::: "memory");
            }
            __syncthreads();
            const float* rows = buf[c & 1];
#else
            copy_chunk_sync(base + (size_t)c * CH_ROWS * NW, buf[0], tid);
            __syncthreads();
            const float* rows = buf[0];
#endif
            // each of the 8 waves reduces 2 rows of this chunk (wave32 shuffles)
#pragma unroll
            for (int rr = 0; rr < 2; ++rr) {
                const int r = 2 * wv + rr;
                const float* row = rows + r * NW;
                float v0 = -INFINITY;
                int   i0 = 0;
#pragma unroll
                for (int k = 0; k < 8; ++k) {
                    int   col = lane + (k << 5);
                    float v   = row[col];
                    if (v > v0) { v0 = v; i0 = col; }
                }
#pragma unroll
                for (int off = 16; off > 0; off >>= 1) {
                    float ov = __shfl_xor(v0, off, 32);
                    int   oi = __shfl_xor(i0, off, 32);
                    amax_combine(v0, i0, ov, oi);
                }
                const int h = c * CH_ROWS + r;           // rows visited in increasing h
                if (v0 > bestv) { bestv = v0; bestr = h; }
            }
            __syncthreads();                              // buffer reuse fence
        }

        if (lane == 0) { wbv[wv] = bestv; wbi[wv] = bestr; }
        __syncthreads();
        if (tid == 0) {
            float pv = wbv[0]; int pr = wbi[0];
#pragma unroll
            for (int w = 1; w < 8; ++w) amax_combine(pv, pr, wbv[w], wbi[w]);
            if (plane == 0) {
                ys_out[bj]    = pr;
                valid_out[bj] = (pv != 0.0f) ? 1 : 0;
            } else {
                pred_out[bj]  = pr;
            }
        }
        __syncthreads();
    }
}

// ---------------------------------------------------------------------------
// Kernel 2 (1 block): recompute head_xs/neck_xs rows, standard[b], final ratio.
// ---------------------------------------------------------------------------
__global__ __launch_bounds__(256) void pckh_final_kernel(
    const float* __restrict__ target,
    const int* __restrict__ ys, const int* __restrict__ pred,
    const int* __restrict__ valid, float* __restrict__ out)
{
    __shared__ int   xs_l[2 * NB];
    __shared__ float std_l[NB];
    __shared__ int   redC[256];
    __shared__ int   redT[256];

    const int tid  = threadIdx.x;
    const int wv   = tid >> 5;
    const int lane = tid & 31;

    // 32 row-argmax tasks: even = head (j=13, row=head_ys), odd = neck (j=1, row=neck_ys)
    for (int task = wv; task < 2 * NB; task += 8) {
        const int b   = task >> 1;
        const int j   = ((task & 1) == 0) ? 13 : 1;
        const int row = ys[b * NJ + j];
        const float* p = target + ((size_t)(b * NJ + j) * NH + row) * NW;

        float v0 = -INFINITY;
        int   i0 = 0;
#pragma unroll
        for (int k = 0; k < 8; ++k) {
            int   col = lane + (k << 5);
            float v   = p[col];
            if (v > v0) { v0 = v; i0 = col; }
        }
#pragma unroll
        for (int off = 16; off > 0; off >>= 1) {
            float ov = __shfl_xor(v0, off, 32);
            int   oi = __shfl_xor(i0, off, 32);
            amax_combine(v0, i0, ov, oi);
        }
        if (lane == 0) xs_l[task] = i0;
    }
    __syncthreads();

    if (tid < NB) {
        const int b  = tid;
        const int hy = ys[b * NJ + 13];
        const int ny = ys[b * NJ + 1];
        const int hx = xs_l[2 * b];
        const int nx = xs_l[2 * b + 1];
        const float dy = (float)(hy - ny);
        const float dx = (float)(hx - nx);
        std_l[b] = sqrtf(dy * dy + dx * dx) * 0.5f;
    }
    __syncthreads();

    int c = 0, t = 0;
    if (tid < NBJ) {
        const int b = tid / NJ;
        const int d = ys[tid] - pred[tid];        // label_xs == predict_xs -> dx term is 0
        const float dist = sqrtf((float)(d * d));
        const int v = valid[tid];
        t = v;
        c = (v && (dist < std_l[b])) ? 1 : 0;
    }
    redC[tid] = c;
    redT[tid] = t;
    __syncthreads();
    for (int s = 128; s > 0; s >>= 1) {
        if (tid < s) { redC[tid] += redC[tid + s]; redT[tid] += redT[tid + s]; }
        __syncthreads();
    }
    if (tid == 0) out[0] = (float)redC[0] / (float)redT[0];
}

// ---------------------------------------------------------------------------
extern "C" void kernel_launch(void* const* d_in, const int* in_sizes, int n_in,
                              void* d_out, int out_size, void* d_ws, size_t ws_size,
                              hipStream_t stream)
{
    const float* x      = (const float*)d_in[0];
    const float* target = (const float*)d_in[1];
    float*       out    = (float*)d_out;

    int* ys    = (int*)d_ws;
    int* pred  = ys + NBJ;
    int* valid = pred + NBJ;

    pckh_stats_kernel<<<NBJ, 256, 0, stream>>>(x, target, ys, pred, valid);
    pckh_final_kernel<<<1, 256, 0, stream>>>(target, ys, pred, valid, out);
}